// TraceLayer_88622355186233
// MI455X (gfx1250) — compile-verified
//
#include <hip/hip_runtime.h>

// Lattice slice: 64 x 64 floats per field, two fields (theta0, theta1) contiguous.
#define TILE     4096          // 64*64
#define THREADS  256           // 8 wave32 waves per workgroup

__global__ __launch_bounds__(THREADS)
void plaq_mean_cos_kernel(const float* __restrict__ x, float* __restrict__ out)
{
    // Single contiguous LDS block so the async copy can cover theta0|theta1 in one sweep.
    __shared__ float smem[2 * TILE];           // 32 KB
    __shared__ float partial[THREADS / 32];    // one partial per wave

    const int tid = threadIdx.x;
    const long long slice = blockIdx.x;                 // 0 .. 4095
    const float* gbase = x + slice * (2LL * TILE);      // theta0 at +0, theta1 at +TILE

    // ---------------- Phase 1: async global -> LDS copy (CDNA5 data mover path) ---
    // 2*TILE floats = 32768 bytes = 2048 x b128 chunks; 8 chunks per lane, coalesced.
    // LDS byte address for async-to-LDS ops is the wave-relative low 32 bits of the
    // flat shared pointer (ISA 10.2).
    const unsigned lds_base = (unsigned)(unsigned long long)(const void*)&smem[0];
    const unsigned long long gaddr_base = (unsigned long long)(const void*)gbase;

#pragma unroll
    for (int k = 0; k < 8; ++k) {
        const unsigned byteOff = (unsigned)((k * THREADS + tid) * 16);
        const unsigned long long ga = gaddr_base + byteOff;   // per-lane 64-bit VGPR pair
        const unsigned la = lds_base + byteOff;               // per-lane LDS byte address
        asm volatile("global_load_async_to_lds_b128 %0, %1, off"
                     :
                     : "v"(la), "v"(ga)
                     : "memory");
    }

    // Wait for this wave's async transfers (ASYNCcnt), then workgroup barrier.
#if __has_builtin(__builtin_amdgcn_s_wait_asynccnt)
    __builtin_amdgcn_s_wait_asynccnt(0);
#else
    asm volatile("s_wait_asynccnt 0" ::: "memory");
#endif
    __syncthreads();

    // ---------------- Phase 2: plaquette + cos, 16 sites per thread ---------------
    // plaq[j,i] = t0[j,i] + t1[j+1,i] - t0[j,i+1] - t1[j,i]   (indices mod 64)
    float acc = 0.0f;
#pragma unroll
    for (int e = 0; e < 16; ++e) {
        const int idx = e * THREADS + tid;       // consecutive lanes -> consecutive banks
        const int j = idx >> 6;
        const int i = idx & 63;
        const float t0  = smem[idx];
        const float t1  = smem[TILE + idx];
        const float t1u = smem[TILE + (((j + 1) & 63) << 6) + i]; // roll(theta1,-1,axis=-2)
        const float t0r = smem[(j << 6) + ((i + 1) & 63)];        // roll(theta0,-1,axis=-1)
        const float p = t0 + t1u - t0r - t1;
        acc += __cosf(p);                         // v_cos_f32 fast path, |p| small
    }

    // ---------------- Phase 3: wave32 + cross-wave reduction ----------------------
#pragma unroll
    for (int off = 16; off > 0; off >>= 1)
        acc += __shfl_down(acc, off, 32);

    const int wave = tid >> 5;
    const int lane = tid & 31;
    if (lane == 0) partial[wave] = acc;
    __syncthreads();

    if (tid == 0) {
        float s = 0.0f;
#pragma unroll
        for (int w = 0; w < THREADS / 32; ++w) s += partial[w];
        out[slice] = s * (1.0f / 4096.0f);
    }
}

extern "C" void kernel_launch(void* const* d_in, const int* in_sizes, int n_in,
                              void* d_out, int out_size, void* d_ws, size_t ws_size,
                              hipStream_t stream)
{
    const float* x = (const float*)d_in[0];
    float* out = (float*)d_out;

    // slices = 128*32 = total elements / (2*64*64)
    const int slices = in_sizes[0] / (2 * TILE);

    plaq_mean_cos_kernel<<<slices, THREADS, 0, stream>>>(x, out);
}